// AgentDetectionLoss_13975823581556
// MI455X (gfx1250) — compile-verified
//
#include <hip/hip_runtime.h>
#include <math.h>

// Problem constants (from reference setup_inputs)
#define NB 128       // batch
#define NQ 900       // queries
#define NG 80        // gt slots
#define NC 92        // NUM_CLASSES+1
#define NO_OBJ 91    // "no object" class
#define HT 256       // hungarian block threads

typedef __attribute__((ext_vector_type(2))) float v2f;
typedef __attribute__((ext_vector_type(8))) float v8f;

// ---------------- wave reductions (wave32) ----------------
__device__ inline float wave_max_f(float x) {
    #pragma unroll
    for (int o = 16; o > 0; o >>= 1) x = fmaxf(x, __shfl_xor(x, o, 32));
    return x;
}
__device__ inline float wave_sum_f(float x) {
    #pragma unroll
    for (int o = 16; o > 0; o >>= 1) x += __shfl_xor(x, o, 32);
    return x;
}

// ---------------- kernel 0: init targets + accumulators ----------------
__global__ void init_kernel(int* __restrict__ target, float* __restrict__ accv, int n) {
    int i = blockIdx.x * blockDim.x + threadIdx.x;
    if (i < n) target[i] = NO_OBJ;
    if (i < 8) accv[i] = 0.0f;
}

// ---------------- kernel 1: rowconst = max + log(sum exp) per (b,q) row ----------------
// one wave per row of 92 logits
__global__ void __launch_bounds__(256) rowconst_kernel(const float* __restrict__ logits,
                                                       float* __restrict__ rowconst, int nrows) {
    int row  = blockIdx.x * 8 + (threadIdx.x >> 5);
    int lane = threadIdx.x & 31;
    if (row >= nrows) return;
    const float* p = logits + (size_t)row * NC;
    float m = -INFINITY;
    for (int j = lane; j < NC; j += 32) m = fmaxf(m, p[j]);
    m = wave_max_f(m);
    float s = 0.0f;
    for (int j = lane; j < NC; j += 32) s += expf(p[j] - m);
    s = wave_sum_f(s);
    if (lane == 0) rowconst[row] = m + logf(s);
}

// ---------------- kernel 2: cost matrix via f32 WMMA (gather-as-onehot-matmul) ----------------
// grid: (8, NB), block 256 (8 waves). wave qt handles queries [16*qt, 16*qt+16).
// M[q,g] = sum_c logits[b,q,c] * onehot(label_g == c)  via V_WMMA_F32_16X16X4_F32, K=92.
// cost_class = rowconst[q] - M[q,g]; store transposed costT[b][g][q] for Hungarian row scans.
__global__ void __launch_bounds__(256) cost_kernel(const float* __restrict__ logits,
                                                   const float* __restrict__ pboxes,
                                                   const int*   __restrict__ labels,
                                                   const float* __restrict__ gboxes,
                                                   const float* __restrict__ rowconst,
                                                   float* __restrict__ costT) {
    const int b    = blockIdx.y;
    const int wave = threadIdx.x >> 5;
    const int lane = threadIdx.x & 31;
    const int qt   = blockIdx.x * 8 + wave;     // 0..56 used (ceil(900/16)=57)
    if (qt >= (NQ + 15) / 16) return;           // wave-uniform exit; WMMA waves keep EXEC all-1
    const int q0 = qt * 16;
    const int m  = lane & 15;                   // A-matrix row within tile
    const int kh = lane >> 4;                   // K half-select (A/B layout per ISA 7.12.2)

    int qa = q0 + m; if (qa > NQ - 1) qa = NQ - 1;   // clamp partial last tile (stores skipped)
    const float* arow = logits + ((size_t)b * NQ + qa) * NC;

    // clipped labels for the 5 g-tiles (one-hot columns this lane owns)
    int lbl[5];
    #pragma unroll
    for (int t = 0; t < 5; ++t) {
        int l = labels[b * NG + t * 16 + m];
        lbl[t] = l < 0 ? 0 : (l > NO_OBJ ? NO_OBJ : l);
    }

    v8f acc[5];
    #pragma unroll
    for (int t = 0; t < 5; ++t)
        #pragma unroll
        for (int e = 0; e < 8; ++e) acc[t][e] = 0.0f;

    // K loop: 92 = 23 steps of K=4. A lane layout: v0=A[m][kh*2], v1=A[m][kh*2+1].
    for (int kk = 0; kk < NC; kk += 4) {
        int ka = kk + kh * 2;
        v2f a; a.x = arow[ka]; a.y = arow[ka + 1];
        #pragma unroll
        for (int t = 0; t < 5; ++t) {
            v2f bb;                              // one-hot B fragment, built in registers
            bb.x = (lbl[t] == ka)     ? 1.0f : 0.0f;
            bb.y = (lbl[t] == ka + 1) ? 1.0f : 0.0f;
            acc[t] = __builtin_amdgcn_wmma_f32_16x16x4_f32(
                false, a, false, bb, (short)0, acc[t], false, false);
        }
    }

    // gt boxes for this lane's 5 columns
    const int n = lane & 15;
    float gcx[5], gcy[5], gw[5], gh[5];
    #pragma unroll
    for (int t = 0; t < 5; ++t) {
        const float* gb = gboxes + ((size_t)b * NG + t * 16 + n) * 4;
        gcx[t] = gb[0]; gcy[t] = gb[1]; gw[t] = gb[2]; gh[t] = gb[3];
    }

    // C/D layout: vgpr r, lane -> row m = r + kh*8, col n = lane&15
    #pragma unroll
    for (int r = 0; r < 8; ++r) {
        int q = q0 + kh * 8 + r;
        if (q >= NQ) continue;
        const float* pb = pboxes + ((size_t)b * NQ + q) * 4;
        float pcx = pb[0], pcy = pb[1], pw = pb[2], ph = pb[3];
        float rc  = rowconst[b * NQ + q];
        float px0 = pcx - 0.5f * pw, py0 = pcy - 0.5f * ph;
        float px1 = pcx + 0.5f * pw, py1 = pcy + 0.5f * ph;
        float parea = (px1 - px0) * (py1 - py0);
        #pragma unroll
        for (int t = 0; t < 5; ++t) {
            float l1 = fabsf(pcx - gcx[t]) + fabsf(pcy - gcy[t]) +
                       fabsf(pw - gw[t]) + fabsf(ph - gh[t]);
            float gx0 = gcx[t] - 0.5f * gw[t], gy0 = gcy[t] - 0.5f * gh[t];
            float gx1 = gcx[t] + 0.5f * gw[t], gy1 = gcy[t] + 0.5f * gh[t];
            float garea = (gx1 - gx0) * (gy1 - gy0);
            float ix0 = fmaxf(px0, gx0), iy0 = fmaxf(py0, gy0);
            float ix1 = fminf(px1, gx1), iy1 = fminf(py1, gy1);
            float iw = fmaxf(ix1 - ix0, 0.0f), ih = fmaxf(iy1 - iy0, 0.0f);
            float inter = iw * ih;
            float uni   = parea + garea - inter;
            float iou   = inter / uni;
            float cx0 = fminf(px0, gx0), cy0 = fminf(py0, gy0);
            float cx1 = fmaxf(px1, gx1), cy1 = fmaxf(py1, gy1);
            float ca  = fmaxf(cx1 - cx0, 0.0f) * fmaxf(cy1 - cy0, 0.0f);
            float giou = iou - (ca - uni) / ca;
            float cost = 5.0f * l1 + (rc - acc[t][r]) - 2.0f * giou;
            int g = t * 16 + n;
            costT[((size_t)b * NG + g) * NQ + q] = cost;
        }
    }
}

// ---------------- kernel 3: per-batch Hungarian (shortest augmenting path) ----------------
// One workgroup per batch; 900-wide scans parallelized over 256 threads with
// wave32 shfl argmin + LDS cross-wave combine. Duals in f64 to match scipy.
__global__ void __launch_bounds__(HT) hungarian_kernel(const float* __restrict__ costT,
                                                       const int*   __restrict__ labels,
                                                       int* __restrict__ target,
                                                       int* __restrict__ matched) {
    const int b   = blockIdx.x;
    const int tid = threadIdx.x;
    const double INFD = 1e300;

    __shared__ double sh_v[NQ];
    __shared__ double sh_short[NQ];
    __shared__ int    sh_path[NQ];
    __shared__ int    sh_r4c[NQ];
    __shared__ unsigned char sh_SC[NQ];
    __shared__ double sh_u[NG];
    __shared__ int    sh_c4r[NG];
    __shared__ unsigned char sh_SR[NG];
    __shared__ double red_v[HT / 32];
    __shared__ int    red_i[HT / 32];
    __shared__ double bc_min;
    __shared__ int    bc_j;

    for (int j = tid; j < NQ; j += HT) { sh_v[j] = 0.0; sh_r4c[j] = -1; }
    for (int i = tid; i < NG; i += HT) { sh_u[i] = 0.0; sh_c4r[i] = -1; }
    __syncthreads();

    for (int i0 = 0; i0 < NG; ++i0) {
        if (labels[b * NG + i0] >= NO_OBJ) continue;   // uniform: skip invalid gt rows
        for (int j = tid; j < NQ; j += HT) { sh_short[j] = INFD; sh_SC[j] = 0; }
        for (int i = tid; i < NG; i += HT) sh_SR[i] = 0;
        __syncthreads();

        double minval = 0.0;
        int i = i0, sink = -1;
        while (sink < 0) {
            if (tid == 0) sh_SR[i] = 1;
            double ui = sh_u[i];
            const float* crow = costT + ((size_t)b * NG + i) * NQ;
            double lv = INFD; int li = 0x7fffffff;
            for (int j = tid; j < NQ; j += HT) {
                if (!sh_SC[j]) {
                    double r = minval + (double)crow[j] - ui - sh_v[j];
                    if (r < sh_short[j]) { sh_short[j] = r; sh_path[j] = i; }
                    double sv = sh_short[j];
                    if (sv < lv || (sv == lv && j < li)) { lv = sv; li = j; }
                }
            }
            #pragma unroll
            for (int o = 16; o > 0; o >>= 1) {        // wave32 argmin (first-index tie-break)
                double ov = __shfl_xor(lv, o, 32);
                int    oi = __shfl_xor(li, o, 32);
                if (ov < lv || (ov == lv && oi < li)) { lv = ov; li = oi; }
            }
            if ((tid & 31) == 0) { red_v[tid >> 5] = lv; red_i[tid >> 5] = li; }
            __syncthreads();
            if (tid == 0) {
                double bv = red_v[0]; int bi = red_i[0];
                for (int w = 1; w < HT / 32; ++w)
                    if (red_v[w] < bv || (red_v[w] == bv && red_i[w] < bi)) { bv = red_v[w]; bi = red_i[w]; }
                bc_min = bv; bc_j = bi; sh_SC[bi] = 1;
            }
            __syncthreads();
            minval = bc_min;
            int j = bc_j;
            int rc = sh_r4c[j];
            if (rc < 0) sink = j; else i = rc;
        }
        // dual update (before augmentation, matching scipy)
        for (int i2 = tid; i2 < NG; i2 += HT) {
            if (i2 == i0) sh_u[i2] += minval;
            else if (sh_SR[i2]) sh_u[i2] += minval - sh_short[sh_c4r[i2]];
        }
        for (int j = tid; j < NQ; j += HT)
            if (sh_SC[j]) sh_v[j] -= minval - sh_short[j];
        __syncthreads();
        if (tid == 0) {                                // augment along path
            int j = sink;
            while (true) {
                int pi = sh_path[j];
                sh_r4c[j] = pi;
                int nj = sh_c4r[pi];
                sh_c4r[pi] = j;
                j = nj;
                if (pi == i0) break;
            }
        }
        __syncthreads();
    }

    for (int i = tid; i < NG; i += HT) {
        int lab = labels[b * NG + i];
        if (lab < NO_OBJ) {
            int q = sh_c4r[i];
            matched[b * NG + i] = q;
            target[b * NQ + q]  = lab;
        } else {
            matched[b * NG + i] = -1;
        }
    }
}

// ---------------- kernel 4: weighted CE sums over all (b,q) ----------------
__global__ void __launch_bounds__(256) ce_kernel(const float* __restrict__ logits,
                                                 const float* __restrict__ rowconst,
                                                 const int*   __restrict__ target,
                                                 float* __restrict__ accv) {
    int idx = blockIdx.x * 256 + threadIdx.x;
    float wn = 0.0f, w = 0.0f;
    if (idx < NB * NQ) {
        int tc = target[idx];
        float nll = rowconst[idx] - logits[(size_t)idx * NC + tc];
        w  = (tc == NO_OBJ) ? 1.0f : 0.1f;
        wn = w * nll;
    }
    __shared__ float s0[256], s1[256];
    s0[threadIdx.x] = wn; s1[threadIdx.x] = w;
    __syncthreads();
    for (int s = 128; s > 0; s >>= 1) {
        if (threadIdx.x < s) { s0[threadIdx.x] += s0[threadIdx.x + s]; s1[threadIdx.x] += s1[threadIdx.x + s]; }
        __syncthreads();
    }
    if (threadIdx.x == 0) { atomicAdd(&accv[0], s0[0]); atomicAdd(&accv[1], s1[0]); }
}

// ---------------- kernel 5: matched-pair L1 / GIoU sums ----------------
__global__ void __launch_bounds__(256) matched_kernel(const float* __restrict__ pboxes,
                                                      const float* __restrict__ gboxes,
                                                      const int*   __restrict__ matched,
                                                      float* __restrict__ accv) {
    int idx = blockIdx.x * 256 + threadIdx.x;
    if (idx >= NB * NG) return;
    int q = matched[idx];
    if (q < 0) return;
    int b = idx / NG;
    const float* pb = pboxes + ((size_t)b * NQ + q) * 4;
    const float* gb = gboxes + (size_t)idx * 4;
    float pcx = pb[0], pcy = pb[1], pw = pb[2], ph = pb[3];
    float gcx = gb[0], gcy = gb[1], gw = gb[2], gh = gb[3];
    float l1 = fabsf(pcx - gcx) + fabsf(pcy - gcy) + fabsf(pw - gw) + fabsf(ph - gh);
    float px0 = pcx - 0.5f * pw, py0 = pcy - 0.5f * ph, px1 = pcx + 0.5f * pw, py1 = pcy + 0.5f * ph;
    float gx0 = gcx - 0.5f * gw, gy0 = gcy - 0.5f * gh, gx1 = gcx + 0.5f * gw, gy1 = gcy + 0.5f * gh;
    float parea = (px1 - px0) * (py1 - py0);
    float garea = (gx1 - gx0) * (gy1 - gy0);
    float iw = fmaxf(fminf(px1, gx1) - fmaxf(px0, gx0), 0.0f);
    float ih = fmaxf(fminf(py1, gy1) - fmaxf(py0, gy0), 0.0f);
    float inter = iw * ih;
    float uni = parea + garea - inter;
    float iou = inter / uni;
    float ca = fmaxf(fmaxf(px1, gx1) - fminf(px0, gx0), 0.0f) *
               fmaxf(fmaxf(py1, gy1) - fminf(py0, gy0), 0.0f);
    float giou = iou - (ca - uni) / ca;
    atomicAdd(&accv[2], l1);
    atomicAdd(&accv[3], 1.0f - giou);
    atomicAdd(&accv[4], 1.0f);
}

// ---------------- kernel 6: finalize ----------------
__global__ void finalize_kernel(const float* __restrict__ accv, float* __restrict__ out) {
    if (threadIdx.x == 0 && blockIdx.x == 0) {
        out[0] = accv[0] / accv[1];
        out[1] = accv[2] / (4.0f * accv[4]);
        out[2] = accv[3] / accv[4];
    }
}

extern "C" void kernel_launch(void* const* d_in, const int* in_sizes, int n_in,
                              void* d_out, int out_size, void* d_ws, size_t ws_size,
                              hipStream_t stream) {
    const float* logits = (const float*)d_in[0];   // [128,900,92]
    const float* pboxes = (const float*)d_in[1];   // [128,900,4]
    const int*   labels = (const int*)  d_in[2];   // [128,80]
    const float* gboxes = (const float*)d_in[3];   // [128,80,4]
    float* out = (float*)d_out;                    // 3 floats

    // workspace layout (~38 MB)
    float* rowconst = (float*)d_ws;                          // NB*NQ
    float* costT    = rowconst + (size_t)NB * NQ;            // NB*NG*NQ (transposed cost)
    int*   target   = (int*)(costT + (size_t)NB * NG * NQ);  // NB*NQ
    int*   matched  = target + (size_t)NB * NQ;              // NB*NG
    float* accv     = (float*)(matched + (size_t)NB * NG);   // 8 accumulators

    // 0) init targets (=no-object) and accumulators
    init_kernel<<<(NB * NQ + 255) / 256, 256, 0, stream>>>(target, accv, NB * NQ);
    // 1) logsumexp rowconst
    rowconst_kernel<<<(NB * NQ + 7) / 8, 256, 0, stream>>>(logits, rowconst, NB * NQ);
    // 2) cost matrix with f32 WMMA
    cost_kernel<<<dim3(8, NB), 256, 0, stream>>>(logits, pboxes, labels, gboxes, rowconst, costT);
    // 3) Hungarian, one workgroup per batch
    hungarian_kernel<<<NB, HT, 0, stream>>>(costT, labels, target, matched);
    // 4) CE sums
    ce_kernel<<<(NB * NQ + 255) / 256, 256, 0, stream>>>(logits, rowconst, target, accv);
    // 5) matched-pair sums
    matched_kernel<<<(NB * NG + 255) / 256, 256, 0, stream>>>(pboxes, gboxes, matched, accv);
    // 6) finalize
    finalize_kernel<<<1, 32, 0, stream>>>(accv, out);
}